// KinematicHungarianMatcher_74131135529390
// MI455X (gfx1250) — compile-verified
//
#include <hip/hip_runtime.h>
#include <hip/hip_bf16.h>
#include <math.h>

constexpr int kB  = 64;
constexpr int kT  = 256;
constexpr int kNQ = 32;
constexpr int kNT = 32;
constexpr int kNR = 31;           // NQ - 1
constexpr int kNP = kB * kT;      // 16384 independent LAP problems

typedef __attribute__((ext_vector_type(2))) float v2f;
typedef __attribute__((ext_vector_type(8))) float v8f;

// ---------------------------------------------------------------------------
// Kernel A: cost matrix. One wave32 per problem, 4 waves per block.
// cost_cls is a real (31x5)x(5x32) matmul -> padded 32x8 x 8x32 via
// V_WMMA_F32_16X16X4_F32 (2 M-tiles x 2 N-tiles x 2 K-chunks = 8 WMMAs).
// ---------------------------------------------------------------------------
__global__ __launch_bounds__(128) void cost_kernel(
    const float* __restrict__ position,   // (B,T,NQ,3)
    const float* __restrict__ velocity,   // (B,T,NQ,3)
    const float* __restrict__ logits,     // (B,T,NQ,5)
    const float* __restrict__ targets,    // (B,T,NT,18)
    float* __restrict__ Cout)             // (B,T,NR,NT)
{
    __shared__ float s_probs[4][32][5];   // row 31 = zero padding
    __shared__ int   s_cls[4][32];        // cls5 per target column
    __shared__ float s_ppos[4][32][3];
    __shared__ float s_pvel[4][32][3];
    __shared__ float s_dot[4][32][32];    // probs . onehot5^T

    const int w    = threadIdx.x >> 5;
    const int lane = threadIdx.x & 31;
    const int prob = blockIdx.x * 4 + w;  // grid sized exactly: 4096*4 = 16384

    // ---- stage predictions: lane == row r (pred index r+1) ----
    {
        const int r = lane;
        if (r < kNR) {
            const float* lg = logits + ((size_t)prob * kNQ + (r + 1)) * 5;
            float x0 = lg[0], x1 = lg[1], x2 = lg[2], x3 = lg[3], x4 = lg[4];
            float m  = fmaxf(fmaxf(fmaxf(x0, x1), fmaxf(x2, x3)), x4);
            float e0 = expf(x0 - m), e1 = expf(x1 - m), e2 = expf(x2 - m);
            float e3 = expf(x3 - m), e4 = expf(x4 - m);
            float inv = 1.0f / (e0 + e1 + e2 + e3 + e4);
            s_probs[w][r][0] = e0 * inv; s_probs[w][r][1] = e1 * inv;
            s_probs[w][r][2] = e2 * inv; s_probs[w][r][3] = e3 * inv;
            s_probs[w][r][4] = e4 * inv;
            const float* pp = position + ((size_t)prob * kNQ + (r + 1)) * 3;
            const float* pv = velocity + ((size_t)prob * kNQ + (r + 1)) * 3;
            s_ppos[w][r][0] = pp[0]; s_ppos[w][r][1] = pp[1]; s_ppos[w][r][2] = pp[2];
            s_pvel[w][r][0] = pv[0]; s_pvel[w][r][1] = pv[1]; s_pvel[w][r][2] = pv[2];
        } else {
            for (int k = 0; k < 5; ++k) s_probs[w][r][k] = 0.0f;
            s_ppos[w][r][0] = s_ppos[w][r][1] = s_ppos[w][r][2] = 0.0f;
            s_pvel[w][r][0] = s_pvel[w][r][1] = s_pvel[w][r][2] = 0.0f;
        }
    }

    // ---- stage targets: lane == column t ----
    float tpx, tpy, tpz, tvx, tvy, tvz;
    {
        const int t = lane;
        const float* tg = targets + ((size_t)prob * kNT + t) * 18;
        tpx = tg[0]; tpy = tg[1]; tpz = tg[2];
        tvx = tg[3]; tvy = tg[4]; tvz = tg[5];
        float o0 = tg[14], o1 = tg[15], o2 = tg[16], o3 = tg[17];
        float s = o0 + o1 + o2 + o3;
        int   idx = 0; float bm = o0;
        if (o1 > bm) { bm = o1; idx = 1; }
        if (o2 > bm) { bm = o2; idx = 2; }
        if (o3 > bm) { bm = o3; idx = 3; }
        s_cls[w][t] = (s > 0.0f) ? (idx + 1) : 0;
    }
    __syncthreads();

    // ---- cost_cls via WMMA: dot[r][t] = sum_k probs[r][k] * [cls5[t]==k] ----
    // A 16x4 f32 layout: v0 = K(0|2), v1 = K(1|3), lane-half selects +2.
    // B 4x16 f32 layout mirrored. D: VGPR g -> M = g + 8*(lane>=16), N = lane%16.
    {
        const int half = lane >> 4;
        const int lm   = lane & 15;
        const int k0lo = 2 * half;          // k-chunk 0: ks {2h, 2h+1}  (all < 5)
        const int k1lo = 4 + 2 * half;      // k-chunk 1: ks {4+2h, 5+2h} (only k=4 real)
        for (int mt = 0; mt < 2; ++mt) {
            const int r = mt * 16 + lm;
            v2f a0, a1;
            a0.x = s_probs[w][r][k0lo];
            a0.y = s_probs[w][r][k0lo + 1];
            a1.x = (k1lo < 5) ? s_probs[w][r][4] : 0.0f;   // k=4 when half==0
            a1.y = 0.0f;                                   // k in {5,7}: padding
            for (int nt = 0; nt < 2; ++nt) {
                const int t  = nt * 16 + lm;
                const int ct = s_cls[w][t];
                v2f b0, b1;
                b0.x = (ct == k0lo)     ? 1.0f : 0.0f;
                b0.y = (ct == k0lo + 1) ? 1.0f : 0.0f;
                b1.x = (ct == k1lo)     ? 1.0f : 0.0f;     // only k=4 can hit
                b1.y = 0.0f;
                v8f acc = {0.f, 0.f, 0.f, 0.f, 0.f, 0.f, 0.f, 0.f};
                acc = __builtin_amdgcn_wmma_f32_16x16x4_f32(
                          false, a0, false, b0, (short)0, acc, false, false);
                acc = __builtin_amdgcn_wmma_f32_16x16x4_f32(
                          false, a1, false, b1, (short)0, acc, false, false);
                #pragma unroll
                for (int g = 0; g < 8; ++g)
                    s_dot[w][mt * 16 + half * 8 + g][t] = acc[g];
            }
        }
    }
    __syncthreads();

    // ---- fuse L1 terms, write C row-major (coalesced across lanes) ----
    {
        const int t = lane;
        float* crow = Cout + (size_t)prob * kNR * kNT;
        for (int r = 0; r < kNR; ++r) {
            float cp = fabsf(s_ppos[w][r][0] - tpx) + fabsf(s_ppos[w][r][1] - tpy) +
                       fabsf(s_ppos[w][r][2] - tpz);
            float cv = fabsf(s_pvel[w][r][0] - tvx) + fabsf(s_pvel[w][r][1] - tvy) +
                       fabsf(s_pvel[w][r][2] - tvz);
            crow[r * kNT + t] = -s_dot[w][r][t] + 5.0f * cp + 2.0f * cv;
        }
    }
}

// ---------------------------------------------------------------------------
// Kernel B: Jonker-Volgenant LAP, one thread per problem, state in LDS
// (transposed [j][tid] layout -> conflict-free across 64 banks).
// Reads C from d_out (fits in the 192MB L2), re-derives valid mask.
// ---------------------------------------------------------------------------
__global__ __launch_bounds__(64) void lap_kernel(
    const float* __restrict__ targets,
    const float* __restrict__ Cmat,
    int* __restrict__ predIdx,
    int* __restrict__ tgtIdx)
{
    const int tid  = threadIdx.x;
    const int prob = blockIdx.x * 64 + tid;   // grid sized exactly: 256*64

    __shared__ float sh_u[32][64];
    __shared__ float sh_v[33][64];
    __shared__ float sh_minv[33][64];
    __shared__ int   sh_p[33][64];
    __shared__ int   sh_way[33][64];
    __shared__ int   sh_used[33][64];

    // valid mask per target column
    unsigned vm = 0;
    for (int t = 0; t < kNT; ++t) {
        const float* oh = targets + ((size_t)prob * kNT + t) * 18 + 14;
        float o0 = oh[0], o1 = oh[1], o2 = oh[2], o3 = oh[3];
        float s = o0 + o1 + o2 + o3;
        int   idx = 0; float bm = o0;
        if (o1 > bm) { bm = o1; idx = 1; }
        if (o2 > bm) { bm = o2; idx = 2; }
        if (o3 > bm) { bm = o3; idx = 3; }
        int cls5 = (s > 0.0f) ? (idx + 1) : 0;
        if (s > 0.0f && cls5 != 1) vm |= (1u << t);
    }

    const float* a = Cmat + (size_t)prob * kNR * kNT;
    const float  INF = 1.0e18f;

    for (int j = 0; j <= 32; ++j) { sh_v[j][tid] = 0.0f; sh_p[j][tid] = 0; }
    for (int i = 0; i < 32; ++i)  sh_u[i][tid] = 0.0f;

    for (int i = 1; i <= kNR; ++i) {
        sh_p[0][tid] = i;
        for (int j = 0; j <= 32; ++j) {
            sh_minv[j][tid] = INF; sh_used[j][tid] = 0; sh_way[j][tid] = 0;
        }
        int j0 = 0;
        while (sh_p[j0][tid] != 0) {
            sh_used[j0][tid] = 1;
            const int   i0  = sh_p[j0][tid];
            const float ui0 = sh_u[i0][tid];
            float delta = INF; int j1 = 0;
            for (int j = 1; j <= 32; ++j) {
                if (!sh_used[j][tid]) {
                    float cij = ((vm >> (j - 1)) & 1u) ? a[(i0 - 1) * kNT + (j - 1)]
                                                       : 1.0e6f;
                    float cur = cij - ui0 - sh_v[j][tid];
                    float mv  = sh_minv[j][tid];
                    if (cur < mv) { mv = cur; sh_minv[j][tid] = cur; sh_way[j][tid] = j0; }
                    if (mv < delta) { delta = mv; j1 = j; }
                }
            }
            for (int j = 0; j <= 32; ++j) {
                if (sh_used[j][tid]) {
                    sh_u[sh_p[j][tid]][tid] += delta;
                    sh_v[j][tid] -= delta;
                } else {
                    sh_minv[j][tid] -= delta;
                }
            }
            j0 = j1;
        }
        while (j0 != 0) {
            int j1 = sh_way[j0][tid];
            sh_p[j0][tid] = sh_p[j1][tid];
            j0 = j1;
        }
    }

    // row2col + outputs
    int row2col[32];
    for (int i = 0; i < 32; ++i) row2col[i] = 0;
    for (int j = 1; j <= 32; ++j) row2col[sh_p[j][tid]] = j;
    for (int r = 0; r < kNR; ++r) {
        int  col = row2col[r + 1] - 1;
        bool av  = (col >= 0) && ((vm >> col) & 1u);
        predIdx[(size_t)prob * kNR + r] = av ? (r + 1) : -1;
        tgtIdx [(size_t)prob * kNR + r] = av ? col     : -1;
    }
}

extern "C" void kernel_launch(void* const* d_in, const int* in_sizes, int n_in,
                              void* d_out, int out_size, void* d_ws, size_t ws_size,
                              hipStream_t stream) {
    (void)in_sizes; (void)n_in; (void)out_size; (void)d_ws; (void)ws_size;
    const float* position = (const float*)d_in[0];
    const float* velocity = (const float*)d_in[1];
    const float* logits   = (const float*)d_in[2];
    const float* targets  = (const float*)d_in[3];

    float* Cout = (float*)d_out;                                   // NP*31*32 floats
    int*   predIdx = (int*)((float*)d_out + (size_t)kNP * kNR * kNT);
    int*   tgtIdx  = predIdx + (size_t)kNP * kNR;

    cost_kernel<<<kNP / 4, 128, 0, stream>>>(position, velocity, logits, targets, Cout);
    lap_kernel<<<kNP / 64, 64, 0, stream>>>(targets, Cout, predIdx, tgtIdx);
}